// downprompt_61478161875367
// MI455X (gfx1250) — compile-verified
//
#include <hip/hip_runtime.h>
#include <math.h>

// -------------------------------------------------------------------------
// downprompt fused kernel for MI455X (gfx1250, wave32)
//
// out[g, f] = sum_{n in graph g} elu( coef[f] * seq[n, f] )
// coef[f]   = df0 * (1 + sum_k wp[k]*prompt[k,f]) + df1 * dp[f]
//
// Memory-bound: 512 MB read @ 23.3 TB/s -> ~22 us floor.
// Strategy: non-temporal coalesced b128 streaming loads, register
// accumulation with dual accumulators, fused coef computation, 2-barrier
// shuffle-based prefix scan for segment offsets, gfx1250 prefetch lookahead.
// -------------------------------------------------------------------------

#define FT 128            // feature width (reference constant)
#define ROWS_PER_TILE 8   // waves per block; one wave covers one row
#define BLOCK_THREADS (ROWS_PER_TILE * 32)

typedef float v4f __attribute__((ext_vector_type(4)));

// ---- Kernel 1: exclusive prefix scan of graph_len (int64) ----------------
// 1024 threads = 32 waves. Intra-wave inclusive scan via shuffles (wave32),
// 32 wave partials scanned by wave 0 through LDS. Only two barriers.
__global__ void scan_kernel(const long long* __restrict__ len,  // [n]
                            long long* __restrict__ offs,       // [n]
                            int n)
{
    const int t    = threadIdx.x;
    const int lane = t & 31;
    const int wave = t >> 5;
    const int per  = (n + 1023) >> 10;   // elements per thread
    const int lo   = t * per;

    // serial sum of this thread's chunk
    long long s = 0;
    for (int i = 0; i < per; ++i) {
        int idx = lo + i;
        if (idx < n) s += len[idx];
    }

    // inclusive scan across the wave (wave32 shuffles, no barriers)
    long long v = s;
    #pragma unroll
    for (int d = 1; d < 32; d <<= 1) {
        long long u = __shfl_up(v, d, 32);
        if (lane >= d) v += u;
    }

    __shared__ long long wsum[32];
    if (lane == 31) wsum[wave] = v;
    __syncthreads();

    if (wave == 0) {
        long long w = wsum[lane];
        #pragma unroll
        for (int d = 1; d < 32; d <<= 1) {
            long long u = __shfl_up(w, d, 32);
            if (lane >= d) w += u;
        }
        wsum[lane] = w;
    }
    __syncthreads();

    const long long wave_prefix = (wave == 0) ? 0 : wsum[wave - 1];
    long long run = wave_prefix + (v - s);   // exclusive prefix of this chunk

    for (int i = 0; i < per; ++i) {
        int idx = lo + i;
        if (idx < n) {
            offs[idx] = run;
            run += len[idx];
        }
    }
}

// ---- ELU helper ----------------------------------------------------------
__device__ __forceinline__ float elu_acc(float c, float x)
{
    float y = c * x;
    return (y > 0.0f) ? y : expm1f(y);
}

// ---- Kernel 2: fused coef + elu + segment-sum pooling --------------------
// One block per graph. Thread layout: qf = lane (feature quad 0..31),
// r = wave id (row-group 0..7). Each wave covers one full 512-byte row per
// step as 32 coalesced non-temporal float4 loads (global_load_b128, TH=NT).
// coef is recomputed per lane from L2-broadcast params (4 b128 loads + ~20
// VALU ops) -- cheaper than a separate kernel + memory round trip.
// Accumulate in registers (unroll x2, dual accumulators), reduce the 8 wave
// partials through 4 KB of LDS, wave 0 stores the 128-float result row.
__global__ __launch_bounds__(BLOCK_THREADS)
void pool_kernel(const float* __restrict__ seq,        // [total_nodes, FT]
                 const long long* __restrict__ len,    // [n_graphs]
                 const long long* __restrict__ offs,   // [n_graphs]
                 const float* __restrict__ prompt,     // [3, FT]
                 const float* __restrict__ wp,         // [1, 3]
                 const float* __restrict__ dp,         // [1, FT]
                 const float* __restrict__ df,         // [1, 2]
                 float* __restrict__ out,              // [n_graphs, FT]
                 int n_graphs)
{
    const int g = blockIdx.x;
    if (g >= n_graphs) return;

    const int qf = threadIdx.x & 31;   // feature quad (lane)
    const int r  = threadIdx.x >> 5;   // wave id / row-group

    // ---- per-lane coefficient quad (uniform params broadcast via L2) ----
    const float wp0 = wp[0], wp1 = wp[1], wp2 = wp[2];
    const float df0 = df[0], df1 = df[1];
    const v4f p0 = ((const v4f*)(prompt        ))[qf];
    const v4f p1 = ((const v4f*)(prompt +   FT ))[qf];
    const v4f p2 = ((const v4f*)(prompt + 2*FT ))[qf];
    const v4f dq = ((const v4f*)(dp            ))[qf];

    v4f c;
    c.x = df0 * (1.0f + wp0 * p0.x + wp1 * p1.x + wp2 * p2.x) + df1 * dq.x;
    c.y = df0 * (1.0f + wp0 * p0.y + wp1 * p1.y + wp2 * p2.y) + df1 * dq.y;
    c.z = df0 * (1.0f + wp0 * p0.z + wp1 * p1.z + wp2 * p2.z) + df1 * dq.z;
    c.w = df0 * (1.0f + wp0 * p0.w + wp1 * p1.w + wp2 * p2.w) + df1 * dq.w;

    const long long start = offs[g];
    const long long end   = start + len[g];

    v4f acc0 = {0.0f, 0.0f, 0.0f, 0.0f};
    v4f acc1 = {0.0f, 0.0f, 0.0f, 0.0f};

    long long row = start + r;

    // main loop: two rows in flight per thread (2 outstanding b128 NT loads)
    for (; row + ROWS_PER_TILE < end; row += 2 * ROWS_PER_TILE) {
        const v4f* ra = (const v4f*)(seq + row * (long long)FT);
        const v4f* rb = (const v4f*)(seq + (row + ROWS_PER_TILE) * (long long)FT);

        // gfx1250 speculative prefetch ~8 tiles (64 rows) ahead of this wave's stream
        __builtin_prefetch((const void*)(seq
            + (row + 8LL * ROWS_PER_TILE) * (long long)FT
            + (long long)qf * 4), 0, 1);

        const v4f va = __builtin_nontemporal_load(ra + qf);
        const v4f vb = __builtin_nontemporal_load(rb + qf);

        acc0.x += elu_acc(c.x, va.x);
        acc0.y += elu_acc(c.y, va.y);
        acc0.z += elu_acc(c.z, va.z);
        acc0.w += elu_acc(c.w, va.w);
        acc1.x += elu_acc(c.x, vb.x);
        acc1.y += elu_acc(c.y, vb.y);
        acc1.z += elu_acc(c.z, vb.z);
        acc1.w += elu_acc(c.w, vb.w);
    }
    // tail: at most one remaining row for this wave
    if (row < end) {
        const v4f* ra = (const v4f*)(seq + row * (long long)FT);
        const v4f va  = __builtin_nontemporal_load(ra + qf);
        acc0.x += elu_acc(c.x, va.x);
        acc0.y += elu_acc(c.y, va.y);
        acc0.z += elu_acc(c.z, va.z);
        acc0.w += elu_acc(c.w, va.w);
    }

    acc0.x += acc1.x; acc0.y += acc1.y; acc0.z += acc1.z; acc0.w += acc1.w;

    // ---- cross-wave reduction: 8 partial float4s per feature quad -------
    __shared__ v4f red[ROWS_PER_TILE][32];
    red[r][qf] = acc0;
    __syncthreads();

    if (r == 0) {
        v4f s = red[0][qf];
        #pragma unroll
        for (int k = 1; k < ROWS_PER_TILE; ++k) {
            v4f t = red[k][qf];
            s.x += t.x; s.y += t.y; s.z += t.z; s.w += t.w;
        }
        ((v4f*)(out + (long long)g * FT))[qf] = s;
    }
}

// -------------------------------------------------------------------------
extern "C" void kernel_launch(void* const* d_in, const int* in_sizes, int n_in,
                              void* d_out, int out_size, void* d_ws, size_t ws_size,
                              hipStream_t stream)
{
    const float*     seq    = (const float*)d_in[0];      // [N, 128] fp32
    const long long* glen   = (const long long*)d_in[1];  // [B] int64
    const float*     prompt = (const float*)d_in[2];      // [3, 128]
    const float*     wp     = (const float*)d_in[3];      // [1, 3]
    const float*     dp     = (const float*)d_in[4];      // [1, 128]
    const float*     df     = (const float*)d_in[5];      // [1, 2]
    float*           out    = (float*)d_out;              // [B, 128]

    const int n_graphs = in_sizes[1];

    // workspace: per-graph exclusive row offsets (int64)
    long long* offs = (long long*)d_ws;

    scan_kernel<<<1, 1024, 0, stream>>>(glen, offs, n_graphs);
    pool_kernel<<<n_graphs, BLOCK_THREADS, 0, stream>>>(seq, glen, offs,
                                                        prompt, wp, dp, df,
                                                        out, n_graphs);
}